// SimplifiedRNNTLoss_4956392259586
// MI455X (gfx1250) — compile-verified
//
#include <hip/hip_runtime.h>
#include <hip/hip_bf16.h>
#include <math.h>

typedef __attribute__((ext_vector_type(2))) float v2f;
typedef __attribute__((ext_vector_type(8))) float v8f;

constexpr int B  = 8;
constexpr int T  = 256;
constexpr int U  = 64;
constexpr int U1 = 65;
constexpr int V  = 1024;
constexpr int ROWS = B * T * U1;   // 133120 rows of length V
#define NEGV (-1e30f)

// ---------------------------------------------------------------------------
// Phase 1: streaming logsumexp over V per (b,t,u) row.
// 16 rows per wave; Σ exp(x) accumulated with V_WMMA_F32_16X16X4_F32 and an
// all-ones B matrix (D[m][n] = Σ_k A[m][k] + C[m][n]) -> full f32 row sums.
// A-matrix 16x4 f32 layout: lanes 0-15 supply K=0,1 ; lanes 16-31 supply K=2,3.
// exp via fast path (v_mul + v_exp_f32): inputs are plain logits, no range
// guarding needed, keeps the loop memory-bound (545 MB / 23.3 TB/s ≈ 23 us).
// ---------------------------------------------------------------------------
__global__ __launch_bounds__(256) void rnnt_lse_kernel(
    const float* __restrict__ logits,        // (B,T,U1,V)
    const int*   __restrict__ targets,       // (B,U)
    const int*   __restrict__ target_lengths,// (B)
    float* __restrict__ blankp,              // (B,T,U1)
    float* __restrict__ yout)                // (B,T,U)
{
  const int lane = threadIdx.x & 31;
  const int wave = blockIdx.x * 8 + (threadIdx.x >> 5);
  const int rowBase = wave * 16;
  const int half = lane >> 4;        // 0: K=0,1   1: K=2,3
  const int m    = lane & 15;        // row within 16-row tile
  const size_t row = (size_t)rowBase + m;
  const float* __restrict__ rp = logits + row * (size_t)V + half * 2;

  v8f c = {0.f, 0.f, 0.f, 0.f, 0.f, 0.f, 0.f, 0.f};
  v2f bones; bones.x = 1.0f; bones.y = 1.0f;

  #pragma unroll 8
  for (int i = 0; i < V / 4; ++i) {
    v2f x = *(const v2f*)(rp + i * 4);
    v2f a;
    a.x = __expf(x.x);               // v_mul_f32 + v_exp_f32
    a.y = __expf(x.y);
    // D = A(16x4) * ones(4x16) + C  ->  row-sum accumulate in f32
    c = __builtin_amdgcn_wmma_f32_16x16x4_f32(
            false, a, false, bones, (short)0, c, false, false);
  }

  // C/D layout: VGPR j holds row j (lanes 0-15) or row j+8 (lanes 16-31),
  // replicated across columns. Lanes 0-7 take rows 0-7, lanes 16-23 rows 8-15.
  float s = c[0];
  #pragma unroll
  for (int j = 1; j < 8; ++j)
    if ((lane & 7) == j) s = c[j];

  if ((lane & 15) < 8) {
    const int rlocal = (lane & 7) + (half << 3);
    const size_t r = (size_t)rowBase + rlocal;
    const float lse = logf(s);
    const float* __restrict__ rpe = logits + r * (size_t)V;
    const int q = (int)(r / U1);          // b*T + t
    const int u = (int)(r - (size_t)q * U1);
    const int b = q >> 8;                 // T = 256
    blankp[r] = rpe[0] - lse;             // BLANK = 0
    if (u < U) {
      const int lab = targets[b * U + u];
      const float yv = (u < target_lengths[b]) ? (rpe[lab] - lse) : NEGV;
      yout[(size_t)q * U + u] = yv;
    }
  }
}

// ---------------------------------------------------------------------------
// Phase 2: alpha recursion. One wave32 per batch, 3 u-slots per lane
// (u = 3*lane + k, u in [0,65)). Per t: exclusive add-scan for c, inclusive
// logaddexp scan for z — all intra-wave shuffles, no barriers.
// Precise expf/log1pf here: errors accumulate over ~256 sequential steps and
// this phase is latency-bound, not throughput-bound.
// ---------------------------------------------------------------------------
__device__ __forceinline__ float lae(float a, float b) {
  float mm = fmaxf(a, b);
  float d  = fabsf(a - b);
  return mm + log1pf(expf(-d));   // expf(-huge) -> 0, log1p(0)=0 : safe
}

__device__ __forceinline__ void excl_cumsum3(int lane, float y0, float y1, float y2,
                                             float& c0, float& c1, float& c2) {
  float s0 = y0, s1 = y0 + y1, s2 = s1 + y2;
  float x = s2;
  #pragma unroll
  for (int d = 1; d < 32; d <<= 1) {
    float t = __shfl_up(x, d, 32);
    if (lane >= d) x += t;
  }
  float pre = __shfl_up(x, 1, 32);
  if (lane == 0) pre = 0.0f;
  c0 = pre; c1 = pre + s0; c2 = pre + s1;
}

__device__ __forceinline__ void incl_logcumsum3(int lane, float w0, float w1, float w2,
                                                float& z0, float& z1, float& z2) {
  float l0 = w0, l1 = lae(w0, w1), l2 = lae(l1, w2);
  float x = l2;
  #pragma unroll
  for (int d = 1; d < 32; d <<= 1) {
    float t = __shfl_up(x, d, 32);
    if (lane >= d) x = lae(x, t);
  }
  float pre = __shfl_up(x, 1, 32);
  if (lane == 0) pre = NEGV;
  z0 = lae(pre, l0); z1 = lae(pre, l1); z2 = lae(pre, l2);
}

__global__ __launch_bounds__(32) void rnnt_alpha_kernel(
    const float* __restrict__ blankp,        // (B,T,U1)
    const float* __restrict__ yout,          // (B,T,U)
    const int*   __restrict__ input_lengths,
    const int*   __restrict__ target_lengths,
    float* __restrict__ ll)                  // (B)
{
  const int b    = blockIdx.x;
  const int lane = threadIdx.x;
  const int il = input_lengths[b];
  const int tl = target_lengths[b];
  const int u0 = lane * 3, u1v = u0 + 1, u2v = u0 + 2;
  const float* __restrict__ bp = blankp + (size_t)b * T * U1;
  const float* __restrict__ yp = yout   + (size_t)b * T * U;

  // t = 0 : alpha0[u] = (u <= tl) ? exclusive-cumsum(y[0,:])[u] : NEG
  float c0, c1, c2;
  {
    float y0 = (u0  < U) ? yp[u0]  : 0.f;
    float y1 = (u1v < U) ? yp[u1v] : 0.f;
    float y2 = (u2v < U) ? yp[u2v] : 0.f;
    excl_cumsum3(lane, y0, y1, y2, c0, c1, c2);
  }
  float a0 = (u0  <= tl) ? c0 : NEGV;
  float a1 = (u1v <= tl) ? c1 : NEGV;
  float a2 = (u2v <= tl) ? c2 : NEGV;

  float afin = -3.0e38f;
  if (il == 1) {
    if (u0 == tl) afin = a0;
    if (u1v == tl) afin = a1;
    if (u2v == tl) afin = a2;
  }

  for (int t = 1; t < il; ++t) {
    const float b0 = (u0  < U1) ? bp[(t - 1) * U1 + u0]  : 0.f;
    const float b1 = (u1v < U1) ? bp[(t - 1) * U1 + u1v] : 0.f;
    const float b2 = (u2v < U1) ? bp[(t - 1) * U1 + u2v] : 0.f;
    {
      float y0 = (u0  < U) ? yp[t * U + u0]  : 0.f;
      float y1 = (u1v < U) ? yp[t * U + u1v] : 0.f;
      float y2 = (u2v < U) ? yp[t * U + u2v] : 0.f;
      excl_cumsum3(lane, y0, y1, y2, c0, c1, c2);
    }
    float w0 = (u0  < U1) ? (a0 + b0 - c0) : NEGV;
    float w1 = (u1v < U1) ? (a1 + b1 - c1) : NEGV;
    float w2 = (u2v < U1) ? (a2 + b2 - c2) : NEGV;
    float z0, z1, z2;
    incl_logcumsum3(lane, w0, w1, w2, z0, z1, z2);
    a0 = (u0  <= tl) ? (c0 + z0) : NEGV;
    a1 = (u1v <= tl) ? (c1 + z1) : NEGV;
    a2 = (u2v <= tl) ? (c2 + z2) : NEGV;
    if (t == il - 1) {
      if (u0 == tl) afin = a0;
      if (u1v == tl) afin = a1;
      if (u2v == tl) afin = a2;
    }
  }

  #pragma unroll
  for (int d = 16; d >= 1; d >>= 1)
    afin = fmaxf(afin, __shfl_xor(afin, d, 32));

  if (lane == 0)
    ll[b] = afin + bp[(il - 1) * U1 + tl];
}

// ---------------------------------------------------------------------------
// Phase 3: loss = mean(-ll)
// ---------------------------------------------------------------------------
__global__ void rnnt_finalize(const float* __restrict__ ll, float* __restrict__ out) {
  if (threadIdx.x == 0) {
    float s = 0.f;
    #pragma unroll
    for (int i = 0; i < B; ++i) s += ll[i];
    out[0] = -s / (float)B;
  }
}

// ---------------------------------------------------------------------------
extern "C" void kernel_launch(void* const* d_in, const int* in_sizes, int n_in,
                              void* d_out, int out_size, void* d_ws, size_t ws_size,
                              hipStream_t stream) {
  const float* logits         = (const float*)d_in[0];
  const int*   targets        = (const int*)d_in[1];
  const int*   input_lengths  = (const int*)d_in[2];
  const int*   target_lengths = (const int*)d_in[3];
  float* out = (float*)d_out;

  float* blankp = (float*)d_ws;            // ROWS floats
  float* yout   = blankp + ROWS;           // B*T*U floats
  float* ll     = yout + (size_t)B * T * U;// B floats

  // 16 rows per wave, 8 waves per block -> ROWS/128 blocks (133120/128 = 1040)
  rnnt_lse_kernel<<<ROWS / 128, 256, 0, stream>>>(logits, targets, target_lengths,
                                                  blankp, yout);
  rnnt_alpha_kernel<<<B, 32, 0, stream>>>(blankp, yout, input_lengths,
                                          target_lengths, ll);
  rnnt_finalize<<<1, 32, 0, stream>>>(ll, out);
}